// GraphSAGE_57647051047656
// MI455X (gfx1250) — compile-verified
//
#include <hip/hip_runtime.h>

// GraphSAGE 2-layer inference for MI455X (gfx1250, wave32, WMMA).
//
// Phase 1: wave-per-edge scatter (coalesced 512B row gather + L2-resident f32 atomics)
// Phase 2: dense layer via V_WMMA_F32_16X16X4_F32 (f32 WMMA, mean-scale folded into A)
// Phase 3/4: same for layer 1.

typedef __attribute__((ext_vector_type(2))) float v2f;
typedef __attribute__((ext_vector_type(8))) float v8f;

#define N0_C 2000000
#define N1_C 200000
#define N2_C 16384
#define DIN_C 128
#define DHID_C 32
#define DOUT_C 64

// -------- workspace layout (bytes) --------
// agg0 [N1,128] f32 | cnt0 [N1] f32 | agg1 [N2,32] f32 | cnt1 [N2] f32 | h [N1,32] f32
#define AGG0_OFF  0ULL
#define AGG0_BYTES (unsigned long long)(N1_C * DIN_C * 4)            // 102,400,000
#define CNT0_OFF  (AGG0_OFF + AGG0_BYTES)
#define CNT0_BYTES (unsigned long long)(N1_C * 4)                    // 800,000
#define AGG1_OFF  (CNT0_OFF + CNT0_BYTES)
#define AGG1_BYTES (unsigned long long)(N2_C * DHID_C * 4)           // 2,097,152
#define CNT1_OFF  (AGG1_OFF + AGG1_BYTES)
#define CNT1_BYTES (unsigned long long)(N2_C * 4)                    // 65,536
#define H_OFF     (CNT1_OFF + CNT1_BYTES)
#define H_BYTES   (unsigned long long)(N1_C * DHID_C * 4)            // 25,600,000
#define ZERO_BYTES (H_OFF)   // agg0+cnt0+agg1+cnt1 are contiguous -> one memset

// ---------------- Phase 1: layer-0 edge scatter ----------------
// One wave32 per edge. Lane reads 4 consecutive floats (float4) of the 128-float
// source row -> the wave issues one fully coalesced 512B read, then 4 atomic
// f32 adds per lane into the destination accumulator (L2-resident, 102MB < 192MB L2).
__global__ void sage_scatter0(const float* __restrict__ x,
                              const int* __restrict__ src,
                              const int* __restrict__ dst,
                              float* __restrict__ agg0,
                              float* __restrict__ cnt0,
                              int e0) {
    int wave = (int)((blockIdx.x * blockDim.x + threadIdx.x) >> 5);
    int lane = threadIdx.x & 31;
    if (wave >= e0) return;                 // wave-uniform
    int s = src[wave];
    int d = dst[wave];
    const float4* xr = (const float4*)(x + (size_t)s * DIN_C);
    float4 v = xr[lane];
    float* ar = agg0 + (size_t)d * DIN_C + (size_t)lane * 4;
    atomicAdd(ar + 0, v.x);
    atomicAdd(ar + 1, v.y);
    atomicAdd(ar + 2, v.z);
    atomicAdd(ar + 3, v.w);
    if (lane == 0) atomicAdd(cnt0 + d, 1.0f);
}

// ---------------- Phase 3: layer-1 edge scatter ----------------
// One wave32 per edge, one float per lane (32-float rows).
__global__ void sage_scatter1(const float* __restrict__ h,
                              const int* __restrict__ src,
                              const int* __restrict__ dst,
                              float* __restrict__ agg1,
                              float* __restrict__ cnt1,
                              int e1) {
    int wave = (int)((blockIdx.x * blockDim.x + threadIdx.x) >> 5);
    int lane = threadIdx.x & 31;
    if (wave >= e1) return;                 // wave-uniform
    int s = src[wave];
    int d = dst[wave];
    float v = h[(size_t)s * DHID_C + lane];
    atomicAdd(agg1 + (size_t)d * DHID_C + lane, v);
    if (lane == 0) atomicAdd(cnt1 + d, 1.0f);
}

// ---------------- Phase 2: layer-0 dense (WMMA f32 16x16x4) ----------------
// One wave computes a 16-row x 32-col tile of h.
//   h = relu( (agg0/cnt) @ W1l^T + x @ W1r^T )
// A (16x4 f32, 2 VGPR):  lane L holds A[L%16][k + (L/16)*2 + {0,1}]
// B (4x16 f32, 2 VGPR):  lane L holds B[k + (L/16)*2 + {0,1}][L%16] = W[n][k]
// C/D (16x16 f32, 8 VGPR): vgpr j, lanes 0-15 -> M=j, lanes 16-31 -> M=j+8, N=L%16
__global__ void sage_l0_dense(const float* __restrict__ agg0,
                              const float* __restrict__ cnt0,
                              const float* __restrict__ x,
                              const float* __restrict__ W1l,
                              const float* __restrict__ W1r,
                              float* __restrict__ h,
                              int n1) {
    int wave = (int)((blockIdx.x * blockDim.x + threadIdx.x) >> 5);
    int lane = threadIdx.x & 31;
    int row0 = wave * 16;
    if (row0 >= n1) return;                 // wave-uniform; n1 % 16 == 0
    int m  = lane & 15;
    int kh = lane >> 4;                     // 0 or 1: which K pair this lane holds
    int row = row0 + m;

    float inv = 1.0f / fmaxf(cnt0[row], 1.0f);   // mean = agg/cnt, folded into A
    const float* aggRow = agg0 + (size_t)row * DIN_C;
    const float* xRow   = x    + (size_t)row * DIN_C;

    v8f c0 = {};   // cols 0..15
    v8f c1 = {};   // cols 16..31

    for (int kk = 0; kk < DIN_C; kk += 4) {
        int k = kk + kh * 2;
        float2 am2 = *(const float2*)(aggRow + k);
        float2 ax2 = *(const float2*)(xRow + k);
        v2f a_m; a_m.x = am2.x * inv; a_m.y = am2.y * inv;
        v2f a_x; a_x.x = ax2.x;       a_x.y = ax2.y;

        float2 t;
        t = *(const float2*)(W1l + (size_t)(0  + m) * DIN_C + k); v2f b0l; b0l.x = t.x; b0l.y = t.y;
        t = *(const float2*)(W1l + (size_t)(16 + m) * DIN_C + k); v2f b1l; b1l.x = t.x; b1l.y = t.y;
        t = *(const float2*)(W1r + (size_t)(0  + m) * DIN_C + k); v2f b0r; b0r.x = t.x; b0r.y = t.y;
        t = *(const float2*)(W1r + (size_t)(16 + m) * DIN_C + k); v2f b1r; b1r.x = t.x; b1r.y = t.y;

        c0 = __builtin_amdgcn_wmma_f32_16x16x4_f32(false, a_m, false, b0l, (short)0, c0, false, false);
        c1 = __builtin_amdgcn_wmma_f32_16x16x4_f32(false, a_m, false, b1l, (short)0, c1, false, false);
        c0 = __builtin_amdgcn_wmma_f32_16x16x4_f32(false, a_x, false, b0r, (short)0, c0, false, false);
        c1 = __builtin_amdgcn_wmma_f32_16x16x4_f32(false, a_x, false, b1r, (short)0, c1, false, false);
    }

    #pragma unroll
    for (int j = 0; j < 8; ++j) {
        int orow = row0 + (kh ? (j + 8) : j);
        h[(size_t)orow * DHID_C + 0  + m] = fmaxf(c0[j], 0.0f);
        h[(size_t)orow * DHID_C + 16 + m] = fmaxf(c1[j], 0.0f);
    }
}

// ---------------- Phase 4: layer-1 dense (WMMA f32 16x16x4) ----------------
// One wave computes a 16-row x 64-col tile of out.
//   out = (agg1/cnt) @ W2l^T + h @ W2r^T     (no relu)
__global__ void sage_l1_dense(const float* __restrict__ agg1,
                              const float* __restrict__ cnt1,
                              const float* __restrict__ h,
                              const float* __restrict__ W2l,
                              const float* __restrict__ W2r,
                              float* __restrict__ out,
                              int n2) {
    int wave = (int)((blockIdx.x * blockDim.x + threadIdx.x) >> 5);
    int lane = threadIdx.x & 31;
    int row0 = wave * 16;
    if (row0 >= n2) return;                 // wave-uniform; n2 % 16 == 0
    int m  = lane & 15;
    int kh = lane >> 4;
    int row = row0 + m;

    float inv = 1.0f / fmaxf(cnt1[row], 1.0f);
    const float* aggRow = agg1 + (size_t)row * DHID_C;
    const float* hRow   = h    + (size_t)row * DHID_C;

    v8f c[4] = {{}, {}, {}, {}};            // 4 column tiles of 16 -> 64 cols

    for (int kk = 0; kk < DHID_C; kk += 4) {
        int k = kk + kh * 2;
        float2 am2 = *(const float2*)(aggRow + k);
        float2 ah2 = *(const float2*)(hRow + k);
        v2f a_m; a_m.x = am2.x * inv; a_m.y = am2.y * inv;
        v2f a_h; a_h.x = ah2.x;       a_h.y = ah2.y;

        #pragma unroll
        for (int t = 0; t < 4; ++t) {
            float2 wl = *(const float2*)(W2l + (size_t)(t * 16 + m) * DHID_C + k);
            float2 wr = *(const float2*)(W2r + (size_t)(t * 16 + m) * DHID_C + k);
            v2f b_l; b_l.x = wl.x; b_l.y = wl.y;
            v2f b_r; b_r.x = wr.x; b_r.y = wr.y;
            c[t] = __builtin_amdgcn_wmma_f32_16x16x4_f32(false, a_m, false, b_l, (short)0, c[t], false, false);
            c[t] = __builtin_amdgcn_wmma_f32_16x16x4_f32(false, a_h, false, b_r, (short)0, c[t], false, false);
        }
    }

    #pragma unroll
    for (int j = 0; j < 8; ++j) {
        int orow = row0 + (kh ? (j + 8) : j);
        #pragma unroll
        for (int t = 0; t < 4; ++t) {
            out[(size_t)orow * DOUT_C + t * 16 + m] = c[t][j];
        }
    }
}

extern "C" void kernel_launch(void* const* d_in, const int* in_sizes, int n_in,
                              void* d_out, int out_size, void* d_ws, size_t ws_size,
                              hipStream_t stream) {
    const float* x   = (const float*)d_in[0];
    const float* W1l = (const float*)d_in[1];
    const float* W1r = (const float*)d_in[2];
    const float* W2l = (const float*)d_in[3];
    const float* W2r = (const float*)d_in[4];
    const int* esrc0 = (const int*)d_in[5];
    const int* edst0 = (const int*)d_in[6];
    const int* esrc1 = (const int*)d_in[7];
    const int* edst1 = (const int*)d_in[8];
    float* out = (float*)d_out;

    const int e0 = in_sizes[5];
    const int e1 = in_sizes[7];

    char* ws = (char*)d_ws;
    float* agg0 = (float*)(ws + AGG0_OFF);
    float* cnt0 = (float*)(ws + CNT0_OFF);
    float* agg1 = (float*)(ws + AGG1_OFF);
    float* cnt1 = (float*)(ws + CNT1_OFF);
    float* h    = (float*)(ws + H_OFF);

    // Zero all accumulators (contiguous region) — graph-capture safe.
    hipMemsetAsync(ws, 0, (size_t)ZERO_BYTES, stream);

    // Phase 1: layer-0 scatter. One wave per edge, 8 waves per 256-thread block.
    {
        int waves = e0;
        int blocks = (waves + 7) / 8;
        sage_scatter0<<<blocks, 256, 0, stream>>>(x, esrc0, edst0, agg0, cnt0, e0);
    }
    // Phase 2: layer-0 dense + relu (WMMA f32).
    {
        int waves = N1_C / 16;              // 12500
        int blocks = (waves + 7) / 8;       // 1563
        sage_l0_dense<<<blocks, 256, 0, stream>>>(agg0, cnt0, x, W1l, W1r, h, N1_C);
    }
    // Phase 3: layer-1 scatter.
    {
        int waves = e1;
        int blocks = (waves + 7) / 8;
        sage_scatter1<<<blocks, 256, 0, stream>>>(h, esrc1, edst1, agg1, cnt1, e1);
    }
    // Phase 4: layer-1 dense (WMMA f32).
    {
        int waves = N2_C / 16;              // 1024
        int blocks = (waves + 7) / 8;       // 128
        sage_l1_dense<<<blocks, 256, 0, stream>>>(agg1, cnt1, h, W2l, W2r, out, N2_C);
    }
}